// GNNForecastor_15375982920128
// MI455X (gfx1250) — compile-verified
//
#include <hip/hip_runtime.h>
#include <hip/hip_bf16.h>

typedef __attribute__((ext_vector_type(2))) float v2f;
typedef __attribute__((ext_vector_type(8))) float v8f;

// ---------------------------------------------------------------------------
// Degree / normalization kernels
// ---------------------------------------------------------------------------
__global__ void k_deg_init(float* __restrict__ deg, int N) {
  int i = blockIdx.x * blockDim.x + threadIdx.x;
  if (i < N) deg[i] = 1.0f;  // self-loop contributes 1 to every node's degree
}

__global__ void k_deg_accum(const long long* __restrict__ ei, float* __restrict__ deg, int E) {
  int e = blockIdx.x * blockDim.x + threadIdx.x;
  if (e < E) {
    int d = (int)ei[(size_t)E + e];
    atomicAdd(&deg[d], 1.0f);
  }
}

__global__ void k_dinv(float* __restrict__ deg, int N) {
  int i = blockIdx.x * blockDim.x + threadIdx.x;
  if (i < N) deg[i] = rsqrtf(deg[i]);  // deg >= 1 always (self-loops)
}

// ---------------------------------------------------------------------------
// GEMM1: h[N,16] = x[N,128] @ W1[128,16]  -- WMMA f32 16x16x4, one wave / 16 rows
// A layout (32-bit 16x4): lanes 0-15 rows M, v0=K+0 v1=K+1; lanes 16-31 v0=K+2 v1=K+3
// C layout: VGPR j, lanes 0-15 -> M=j, lanes 16-31 -> M=j+8; N = lane&15
// ---------------------------------------------------------------------------
__global__ void k_gemm_xw1(const float* __restrict__ x, const float* __restrict__ W,
                           float* __restrict__ h, int N) {
  int wave    = blockIdx.x * (blockDim.x >> 5) + (threadIdx.x >> 5);
  int lane    = threadIdx.x & 31;
  int laneLo  = lane & 15;
  int laneHi  = lane >> 4;
  int rowBase = wave * 16;
  if (rowBase >= N) return;  // wave-uniform exit

  int rowA = rowBase + laneLo;
  if (rowA >= N) rowA = N - 1;  // clamp: keeps EXEC all-ones, masked at store

  v8f acc = {};
  #pragma unroll 4
  for (int k = 0; k < 128; k += 4) {
    v2f a, b;
    const float* xp = x + (size_t)rowA * 128 + k + 2 * laneHi;
    a.x = xp[0];
    a.y = xp[1];
    const float* wp = W + (size_t)(k + 2 * laneHi) * 16 + laneLo;
    b.x = wp[0];
    b.y = wp[16];
    acc = __builtin_amdgcn_wmma_f32_16x16x4_f32(false, a, false, b, (short)0, acc,
                                                false, false);
  }

  float* hp = h + (size_t)(rowBase + 8 * laneHi) * 16 + laneLo;
  if (rowBase + 16 <= N) {           // wave-uniform fast path: unguarded stores
    #pragma unroll
    for (int j = 0; j < 8; ++j) hp[(size_t)j * 16] = acc[j];
  } else {                           // only the single tail wave takes this
    #pragma unroll
    for (int j = 0; j < 8; ++j) {
      int m = rowBase + j + 8 * laneHi;
      if (m < N) h[(size_t)m * 16 + laneLo] = acc[j];
    }
  }
}

// ---------------------------------------------------------------------------
// GEMM2: h2[N,16] = h[N,16] @ W2[16,16]  (K=16 -> 4 WMMA steps)
// ---------------------------------------------------------------------------
__global__ void k_gemm_16x16(const float* __restrict__ hin, const float* __restrict__ W,
                             float* __restrict__ hout, int N) {
  int wave    = blockIdx.x * (blockDim.x >> 5) + (threadIdx.x >> 5);
  int lane    = threadIdx.x & 31;
  int laneLo  = lane & 15;
  int laneHi  = lane >> 4;
  int rowBase = wave * 16;
  if (rowBase >= N) return;

  int rowA = rowBase + laneLo;
  if (rowA >= N) rowA = N - 1;

  v8f acc = {};
  #pragma unroll
  for (int k = 0; k < 16; k += 4) {
    v2f a, b;
    const float* ap = hin + (size_t)rowA * 16 + k + 2 * laneHi;
    a.x = ap[0];
    a.y = ap[1];
    const float* wp = W + (size_t)(k + 2 * laneHi) * 16 + laneLo;
    b.x = wp[0];
    b.y = wp[16];
    acc = __builtin_amdgcn_wmma_f32_16x16x4_f32(false, a, false, b, (short)0, acc,
                                                false, false);
  }

  float* hp = hout + (size_t)(rowBase + 8 * laneHi) * 16 + laneLo;
  if (rowBase + 16 <= N) {
    #pragma unroll
    for (int j = 0; j < 8; ++j) hp[(size_t)j * 16] = acc[j];
  } else {
    #pragma unroll
    for (int j = 0; j < 8; ++j) {
      int m = rowBase + j + 8 * laneHi;
      if (m < N) hout[(size_t)m * 16 + laneLo] = acc[j];
    }
  }
}

// ---------------------------------------------------------------------------
// GEMM3: out[N,128] = h[N,16] @ Wl[16,128] + bl   (8 column blocks of 16)
// ---------------------------------------------------------------------------
__global__ void k_gemm_out(const float* __restrict__ hin, const float* __restrict__ W,
                           const float* __restrict__ bias, float* __restrict__ out, int N) {
  int wave    = blockIdx.x * (blockDim.x >> 5) + (threadIdx.x >> 5);
  int lane    = threadIdx.x & 31;
  int laneLo  = lane & 15;
  int laneHi  = lane >> 4;
  int rowBase = wave * 16;
  if (rowBase >= N) return;

  int rowA = rowBase + laneLo;
  if (rowA >= N) rowA = N - 1;

  v8f acc[8];
  #pragma unroll
  for (int nb = 0; nb < 8; ++nb) acc[nb] = (v8f){};

  #pragma unroll
  for (int k = 0; k < 16; k += 4) {
    v2f a;
    const float* ap = hin + (size_t)rowA * 16 + k + 2 * laneHi;
    a.x = ap[0];
    a.y = ap[1];
    #pragma unroll
    for (int nb = 0; nb < 8; ++nb) {
      v2f b;
      const float* wp = W + (size_t)(k + 2 * laneHi) * 128 + nb * 16 + laneLo;
      b.x = wp[0];
      b.y = wp[128];
      acc[nb] = __builtin_amdgcn_wmma_f32_16x16x4_f32(false, a, false, b, (short)0,
                                                      acc[nb], false, false);
    }
  }

  float* op = out + (size_t)(rowBase + 8 * laneHi) * 128 + laneLo;
  if (rowBase + 16 <= N) {           // wave-uniform fast path
    #pragma unroll
    for (int nb = 0; nb < 8; ++nb) {
      float bv = bias[nb * 16 + laneLo];
      #pragma unroll
      for (int j = 0; j < 8; ++j) op[(size_t)j * 128 + nb * 16] = acc[nb][j] + bv;
    }
  } else {
    #pragma unroll
    for (int nb = 0; nb < 8; ++nb) {
      float bv = bias[nb * 16 + laneLo];
      #pragma unroll
      for (int j = 0; j < 8; ++j) {
        int m = rowBase + j + 8 * laneHi;
        if (m < N) out[(size_t)m * 128 + nb * 16 + laneLo] = acc[nb][j] + bv;
      }
    }
  }
}

// ---------------------------------------------------------------------------
// Aggregation: self-loop seed, edge scatter-add, bias+ReLU
// ---------------------------------------------------------------------------
__global__ void k_agg_self(const float* __restrict__ hin, const float* __restrict__ dinv,
                           float* __restrict__ hout, int N) {
  int t = blockIdx.x * blockDim.x + threadIdx.x;  // N*4 threads, float4 each
  int i = t >> 2, q = t & 3;
  if (i < N) {
    float s = dinv[i] * dinv[i];  // self-loop norm
    float4 v = *(const float4*)(hin + (size_t)i * 16 + q * 4);
    v.x *= s; v.y *= s; v.z *= s; v.w *= s;
    *(float4*)(hout + (size_t)i * 16 + q * 4) = v;
  }
}

__global__ void k_agg_edges(const long long* __restrict__ ei, const float* __restrict__ hin,
                            const float* __restrict__ dinv, float* __restrict__ hout, int E) {
  int t = blockIdx.x * blockDim.x + threadIdx.x;  // E*4 threads, 4 features each
  int e = t >> 2, q = t & 3;
  if (e < E) {
    int s = (int)ei[e];
    int d = (int)ei[(size_t)E + e];
    float nrm = dinv[s] * dinv[d];
    float4 v = *(const float4*)(hin + (size_t)s * 16 + q * 4);
    float* o = hout + (size_t)d * 16 + q * 4;
    atomicAdd(o + 0, v.x * nrm);
    atomicAdd(o + 1, v.y * nrm);
    atomicAdd(o + 2, v.z * nrm);
    atomicAdd(o + 3, v.w * nrm);
  }
}

__global__ void k_bias_relu(const float* __restrict__ hin, const float* __restrict__ b,
                            float* __restrict__ hout, int NH) {
  int t = blockIdx.x * blockDim.x + threadIdx.x;
  if (t < NH) {
    float v = hin[t] + b[t & 15];
    hout[t] = v > 0.0f ? v : 0.0f;
  }
}

// ---------------------------------------------------------------------------
// Orchestration
// ---------------------------------------------------------------------------
extern "C" void kernel_launch(void* const* d_in, const int* in_sizes, int n_in,
                              void* d_out, int out_size, void* d_ws, size_t ws_size,
                              hipStream_t stream) {
  const float*     x   = (const float*)d_in[0];
  const long long* ei  = (const long long*)d_in[1];
  const float*     W1  = (const float*)d_in[2];
  const float*     b1  = (const float*)d_in[3];
  const float*     W2  = (const float*)d_in[4];
  const float*     b2  = (const float*)d_in[5];
  const float*     Wl  = (const float*)d_in[6];
  const float*     bl  = (const float*)d_in[7];
  float*           out = (float*)d_out;

  const int N = in_sizes[0] / 128;  // x is [N, 128]
  const int E = in_sizes[1] / 2;    // edge_index is [2, E]

  float* ws   = (float*)d_ws;
  size_t Na   = ((size_t)N + 3) & ~(size_t)3;  // keep 16B alignment
  float* dinv = ws;                     // [N]
  float* bufA = ws + Na;                // [N,16]
  float* bufB = bufA + (size_t)N * 16;  // [N,16]

  const int T  = 256;
  const int gN  = (N + T - 1) / T;
  const int gE  = (E + T - 1) / T;
  const int gN4 = (N * 4 + T - 1) / T;
  const int gE4 = (E * 4 + T - 1) / T;
  const int gNH = (N * 16 + T - 1) / T;
  const int gW  = ((N + 15) / 16 + 3) / 4;  // waves=ceil(N/16), 4 waves/block

  // deg -> dinv
  k_deg_init <<<gN, T, 0, stream>>>(dinv, N);
  k_deg_accum<<<gE, T, 0, stream>>>(ei, dinv, E);
  k_dinv     <<<gN, T, 0, stream>>>(dinv, N);

  // layer 1: transform, propagate, relu
  k_gemm_xw1 <<<gW, 128, 0, stream>>>(x, W1, bufA, N);
  k_agg_self <<<gN4, T, 0, stream>>>(bufA, dinv, bufB, N);
  k_agg_edges<<<gE4, T, 0, stream>>>(ei, bufA, dinv, bufB, E);
  k_bias_relu<<<gNH, T, 0, stream>>>(bufB, b1, bufA, N * 16);

  // layer 2: transform, propagate, relu
  k_gemm_16x16<<<gW, 128, 0, stream>>>(bufA, W2, bufB, N);
  k_agg_self  <<<gN4, T, 0, stream>>>(bufB, dinv, bufA, N);
  k_agg_edges <<<gE4, T, 0, stream>>>(ei, bufB, dinv, bufA, E);
  k_bias_relu <<<gNH, T, 0, stream>>>(bufA, b2, bufB, N * 16);

  // output head with fused bias
  k_gemm_out<<<gW, 128, 0, stream>>>(bufB, Wl, bl, out, N);
}